// Net_50328426774680
// MI455X (gfx1250) — compile-verified
//
#include <hip/hip_runtime.h>

typedef float v4f __attribute__((ext_vector_type(4)));

#define B_  8
#define H_  512
#define W_  512
#define HW_ (H_ * W_)
#define P_  20          // TL - IL = 29 - 9
#define TL_ 29
#define IL_ 9

// One scan step. Grid is (HW/4/256, B): blockIdx.y is the batch image, so all
// per-image base addresses are wave-uniform (SALU) and per-lane address math is
// pure 32-bit offsets (GVS-mode global loads). 4 pixels/thread, 128-bit IO.
__global__ __launch_bounds__(256) void evo_step_kernel(
    const float* __restrict__ carry_in, int carry_in_stride,   // per-b plane stride
    float* __restrict__ carry_out,                             // (B,H,W) in ws
    const float* __restrict__ motion,                          // (B,2P,H,W)
    const float* __restrict__ intensity,                       // (B,P,H,W)
    float* __restrict__ evo_result,                            // (B,P,H,W)
    float* __restrict__ evo_motion,                            // (B,P,H,W)
    float* __restrict__ out_motion,                            // (B,P,2,H,W) == input layout
    int t)
{
    const int b   = blockIdx.y;                       // wave-uniform batch index
    const int pix = (blockIdx.x * 256 + threadIdx.x) * 4;   // [0, HW), multiple of 4
    const int y   = pix >> 9;                         // W == 512
    const int x   = pix & (W_ - 1);                   // 16B-aligned vector start

    // All of these are scalar (block-uniform) base pointers.
    const float* fxp = motion    + (long)b * (2 * P_ * HW_) + (long)(2 * t)     * HW_ + pix;
    const float* fyp = motion    + (long)b * (2 * P_ * HW_) + (long)(2 * t + 1) * HW_ + pix;
    const float* inp = intensity + (long)b * (P_ * HW_)     + (long)t           * HW_ + pix;
    const float* base = carry_in + (long)b * carry_in_stride;

    // Streaming inputs: non-temporal so they don't evict the carry from L2.
    v4f fx4 = __builtin_nontemporal_load((const v4f*)fxp);
    v4f fy4 = __builtin_nontemporal_load((const v4f*)fyp);
    v4f in4 = __builtin_nontemporal_load((const v4f*)inp);

    v4f nxt4, bil4;
#pragma unroll
    for (int j = 0; j < 4; ++j) {
        float cx = fminf(fmaxf((float)(x + j) + fx4[j], 0.0f), (float)(W_ - 1));
        float cy = fminf(fmaxf((float)y       + fy4[j], 0.0f), (float)(H_ - 1));
        float x0f = floorf(cx);
        float y0f = floorf(cy);
        float wx = cx - x0f;
        float wy = cy - y0f;
        int x0 = (int)x0f;                 // already in [0, 511]
        int y0 = (int)y0f;
        int x1 = min(x0 + 1, W_ - 1);
        int y1 = min(y0 + 1, H_ - 1);

        // 32-bit offsets from a scalar base -> 4 gathers, all L2 hits
        // (carry working set is 8 MB vs 192 MB L2).
        int r0 = y0 << 9;
        int r1 = y1 << 9;
        float v00 = base[r0 + x0];
        float v01 = base[r0 + x1];
        float v10 = base[r1 + x0];
        float v11 = base[r1 + x1];

        float top = fmaf(wx, v01 - v00, v00);
        float bot = fmaf(wx, v11 - v10, v10);
        bil4[j]   = fmaf(wy, bot - top, top);

        // nearest (round-half-even like jnp.round) is always one of the 4 corners
        int xn = min((int)rintf(cx), W_ - 1);
        int yn = min((int)rintf(cy), H_ - 1);
        float vn = (yn > y0) ? ((xn > x0) ? v11 : v10)
                             : ((xn > x0) ? v01 : v00);
        nxt4[j] = vn + in4[j];
    }

    // Carry for next step: regular-temporal store -> stays hot in L2.
    *(v4f*)(carry_out + (long)b * HW_ + pix) = nxt4;

    // Streaming outputs: non-temporal stores.
    float* resp = evo_result + (long)b * (P_ * HW_) + (long)t * HW_ + pix;
    float* bilp = evo_motion + (long)b * (P_ * HW_) + (long)t * HW_ + pix;
    __builtin_nontemporal_store(nxt4 * (1.0f / 128.0f), (v4f*)resp);
    __builtin_nontemporal_store(bil4,                   (v4f*)bilp);
    // Fused motion copy (output 3 has identical memory layout to the input).
    float* omx = out_motion + (long)b * (2 * P_ * HW_) + (long)(2 * t)     * HW_ + pix;
    float* omy = out_motion + (long)b * (2 * P_ * HW_) + (long)(2 * t + 1) * HW_ + pix;
    __builtin_nontemporal_store(fx4, (v4f*)omx);
    __builtin_nontemporal_store(fy4, (v4f*)omy);

    // Warm L2 for the NEXT step's streaming inputs (gfx1250 global_prefetch_b8):
    // this kernel is gather-latency bound, so the prefetches ride for free.
    if (t + 1 < P_) {
        __builtin_prefetch(fxp + 2 * HW_, 0, 1);
        __builtin_prefetch(fyp + 2 * HW_, 0, 1);
        __builtin_prefetch(inp + HW_,     0, 1);
    }
}

extern "C" void kernel_launch(void* const* d_in, const int* in_sizes, int n_in,
                              void* d_out, int out_size, void* d_ws, size_t ws_size,
                              hipStream_t stream) {
    (void)in_sizes; (void)n_in; (void)out_size;

    const float* all_frames = (const float*)d_in[0];  // (8, 29, 512, 512, 1)
    const float* motion     = (const float*)d_in[1];  // (8, 40, 512, 512)
    const float* intensity  = (const float*)d_in[2];  // (8, 20, 512, 512)

    float* evo_result = (float*)d_out;                         // (8, 20, 512, 512)
    float* evo_motion = evo_result + (long)B_ * P_ * HW_;      // (8, 20, 512, 512)
    float* out_motion = evo_motion + (long)B_ * P_ * HW_;      // (8, 20, 2, 512, 512)

    if (ws_size < 2ull * B_ * HW_ * sizeof(float)) return;     // need 16 MB ping-pong
    float* carryA = (float*)d_ws;
    float* carryB = carryA + (long)B_ * HW_;

    // 256 threads (8 wave32s); 256 blocks per image x 8 images.
    const dim3 block(256, 1, 1);
    const dim3 grid(HW_ / 4 / 256, B_, 1);                     // (256, 8)

    // Sequential scan: kernel boundary provides the device-wide ordering
    // each step needs (gathers read arbitrary pixels of the previous carry).
    for (int t = 0; t < P_; ++t) {
        const float* cin = (t == 0)
            ? (all_frames + (long)(IL_ - 1) * HW_)             // frame 8 of each batch
            : ((t & 1) ? carryB : carryA);
        const int stride = (t == 0) ? (TL_ * HW_) : HW_;       // per-b plane stride
        float* cout = (t & 1) ? carryA : carryB;

        evo_step_kernel<<<grid, block, 0, stream>>>(
            cin, stride, cout, motion, intensity,
            evo_result, evo_motion, out_motion, t);
    }
}